// TrainingConstraintWrapper_3427383902410
// MI455X (gfx1250) — compile-verified
//
#include <hip/hip_runtime.h>

typedef float v2f __attribute__((ext_vector_type(2)));
typedef float v8f __attribute__((ext_vector_type(8)));

#define TM      16            // rows per workgroup tile
#define D_      512
#define F_      2048
#define V_      22
#define S_      512
#define XPAD    516           // x-tile row stride in floats (bank-spread pad)
#define HPAD    20            // h-buffer row stride in floats
#define NWAVES  8
#define FPAIRS  8             // 8 pairs of f-slices per wave (16 slices total)

static __device__ __forceinline__ v8f wmma_f32_k4(v2f a, v2f b, v8f c) {
  // D = A(16x4 f32) * B(4x16 f32) + C(16x16 f32) -> v_wmma_f32_16x16x4_f32
  return __builtin_amdgcn_wmma_f32_16x16x4_f32(
      /*neg_a=*/false, a, /*neg_b=*/false, b,
      /*c_mod=*/(short)0, c, /*reuse_a=*/false, /*reuse_b=*/false);
}

// bias + relu + transpose (D-layout -> A-layout via per-wave LDS) + GEMM2 accum
static __device__ __forceinline__ void h_epilogue(
    v8f acc, int f, float* hb, const float* __restrict__ b1,
    const float* __restrict__ W2, int n16, int hi, int n2c, float m2,
    v8f& lacc0, v8f& lacc1)
{
  const float bias = b1[f + n16];
  #pragma unroll
  for (int r = 0; r < 8; ++r) {
    float t = acc[r] + bias;
    acc[r] = t > 0.0f ? t : 0.0f;
  }
  // D-layout: VGPR r, lane -> row (r + 8*hi), col n16
  #pragma unroll
  for (int r = 0; r < 8; ++r)
    hb[(r + 8 * hi) * HPAD + n16] = acc[r];

  // logits(16x32) += h_tile(16x16) @ W2[f:f+16, :]   (DS pipe is in-order per wave)
  #pragma unroll
  for (int kk = 0; kk < 4; ++kk) {
    v2f a = *reinterpret_cast<const v2f*>(hb + n16 * HPAD + kk * 4 + 2 * hi);
    const int k0 = f + kk * 4 + 2 * hi;
    v2f bb0;
    bb0.x = W2[k0 * V_ + n16];
    bb0.y = W2[(k0 + 1) * V_ + n16];
    lacc0 = wmma_f32_k4(a, bb0, lacc0);
    v2f bb1;                       // branch-free tail: clamped index * {0,1} mask
    bb1.x = W2[k0 * V_ + n2c] * m2;
    bb1.y = W2[(k0 + 1) * V_ + n2c] * m2;
    lacc1 = wmma_f32_k4(a, bb1, lacc1);
  }
}

__global__ __launch_bounds__(256) void fused_mlp_constraint_kernel(
    const int*   __restrict__ tokens,  // (B,S)
    const float* __restrict__ z,       // (B,D)
    const float* __restrict__ emb,     // (V,D)
    const float* __restrict__ W1,      // (D,F)
    const float* __restrict__ b1,      // (F)
    const float* __restrict__ W2,      // (F,V)
    const float* __restrict__ b2,      // (V)
    float*       __restrict__ out)     // (B,S,V)
{
  __shared__ __align__(16) float xs[TM * XPAD];            // 16 x 516 fp32
  __shared__ __align__(16) float hbuf[NWAVES][TM * HPAD];  // per-wave h transpose
  __shared__ __align__(16) float lred[TM * 32];            // logits reduce (V pad 32)
  __shared__ int ndig_sh;

  const int tid   = threadIdx.x;
  const int wave  = tid >> 5;
  const int lane  = tid & 31;
  const int hi    = lane >> 4;       // half-wave select (K pairs)
  const int n16   = lane & 15;       // M for A reads / N for B,C,D
  const int row0  = blockIdx.x * TM; // global row base (tile within one batch)
  const int batch = row0 / S_;

  // ---- stage 0: zero reduction buffer, count digits, build x tile ----
  lred[tid] = 0.0f;
  lred[tid + 256] = 0.0f;
  if (tid == 0) ndig_sh = 0;
  __syncthreads();

  {
    int c = 0;
    for (int i = tid; i < S_ - 1; i += 256) {
      int t = tokens[batch * S_ + i];
      c += (t >= 13 && t <= 21) ? 1 : 0;   // '1'..'9'
    }
    if (c) atomicAdd(&ndig_sh, c);
  }

  // x[r][d] = emb[tok[row0+r]][d] + z[batch][d]; vectorized float4
  for (int idx = tid; idx < TM * (D_ / 4); idx += 256) {
    int r = idx >> 7;            // / 128
    int d = (idx & 127) << 2;    // *4
    int tok = tokens[row0 + r];
    const float4 e  = *reinterpret_cast<const float4*>(emb + tok * D_ + d);
    const float4 zv = *reinterpret_cast<const float4*>(z + batch * D_ + d);
    float4 xv = make_float4(e.x + zv.x, e.y + zv.y, e.z + zv.z, e.w + zv.w);
    *reinterpret_cast<float4*>(xs + r * XPAD + d) = xv;
  }
  __syncthreads();

  const int ndig = ndig_sh;
  const int nnon = (S_ - 1) - ndig;

  // hoisted GEMM2 tail mask (lanes with n16+16 >= V load a clamped addr * 0)
  const float m2  = (n16 + 16 < V_) ? 1.0f : 0.0f;
  const int   n2c = (n16 + 16 < V_) ? (n16 + 16) : (V_ - 1);

  // ---- per-wave fused GEMM1 (two f-slices per K-step) -> GEMM2 accumulation ----
  v8f lacc0 = {0.f,0.f,0.f,0.f,0.f,0.f,0.f,0.f};
  v8f lacc1 = {0.f,0.f,0.f,0.f,0.f,0.f,0.f,0.f};
  float* hb = &hbuf[wave][0];
  const float* xrow = xs + n16 * XPAD + 2 * hi;   // A: M=n16, K pair base

  for (int fi = 0; fi < FPAIRS; ++fi) {
    const int f0 = (wave * 2 + fi * 2 * NWAVES) * 16;   // pair {f0, f0+16}

    v8f acc0 = {0.f,0.f,0.f,0.f,0.f,0.f,0.f,0.f};
    v8f acc1 = {0.f,0.f,0.f,0.f,0.f,0.f,0.f,0.f};
    const float* w1p = W1 + (2 * hi) * F_ + f0 + n16;   // B: K row base, N=n16

    #pragma unroll 4
    for (int kk = 0; kk < D_ / 4; ++kk) {
      v2f a = *reinterpret_cast<const v2f*>(xrow + kk * 4);  // shared A load
      const size_t rb = (size_t)(kk * 4) * F_;
      v2f b0;                                  // f-slice 0 columns
      b0.x = w1p[rb];
      b0.y = w1p[rb + F_];
      v2f b1v;                                 // f-slice 1 columns (+64B)
      b1v.x = w1p[rb + 16];
      b1v.y = w1p[rb + F_ + 16];
      acc0 = wmma_f32_k4(a, b0, acc0);
      acc1 = wmma_f32_k4(a, b1v, acc1);
    }

    h_epilogue(acc0, f0,      hb, b1, W2, n16, hi, n2c, m2, lacc0, lacc1);
    h_epilogue(acc1, f0 + 16, hb, b1, W2, n16, hi, n2c, m2, lacc0, lacc1);
  }

  // ---- cross-wave reduce ----
  #pragma unroll
  for (int r = 0; r < 8; ++r) {
    atomicAdd(&lred[(r + 8 * hi) * 32 + n16], lacc0[r]);
    atomicAdd(&lred[(r + 8 * hi) * 32 + 16 + n16], lacc1[r]);
  }
  __syncthreads();

  // ---- bias2 + constraint mask + store ----
  for (int idx = tid; idx < TM * V_; idx += 256) {
    int m = idx / V_;
    int v = idx - m * V_;
    float dd = (v <= 2 || v >= 13) ? 1.0f : 0.0f;   // D_DIGIT
    float dn = (v <= 2 || v == 12) ? 1.0f : 0.0f;   // D_NONDIGIT
    float mask = (float)ndig * dd + (float)nnon * dn;
    out[(size_t)(row0 + m) * V_ + v] = lred[m * 32 + v] + b2[v] - 0.5f * mask;
  }
}

extern "C" void kernel_launch(void* const* d_in, const int* in_sizes, int n_in,
                              void* d_out, int out_size, void* d_ws, size_t ws_size,
                              hipStream_t stream) {
  (void)in_sizes; (void)n_in; (void)d_ws; (void)ws_size; (void)out_size;
  const int*   tokens = (const int*)  d_in[0];
  const float* z      = (const float*)d_in[1];
  const float* emb    = (const float*)d_in[2];
  const float* W1     = (const float*)d_in[3];
  const float* b1     = (const float*)d_in[4];
  const float* W2     = (const float*)d_in[5];
  const float* b2     = (const float*)d_in[6];
  float*       out    = (float*)d_out;

  const int n_row_tiles = (32 * S_) / TM;  // 1024
  fused_mlp_constraint_kernel<<<n_row_tiles, 256, 0, stream>>>(
      tokens, z, emb, W1, b1, W2, b2, out);
}